// ExpandedSiameseConcat_76132590289284
// MI455X (gfx1250) — compile-verified
//
#include <hip/hip_runtime.h>
#include <hip/hip_bf16.h>
#include <stdint.h>

// ---------------------------------------------------------------------------
// ExpandedSiameseConcat on MI455X (gfx1250)
//
// Zero-FLOP gather/concat => store-bandwidth bound: 528 MB of output stores,
// ~23 us floor at 23.3 TB/s. curr (4 MB) lives in the 192 MB L2, so reads are
// ~free after first touch. Strategy: drive the Tensor Data Mover. One wave per
// anchor stages the anchor row once, then per group of 4 output rows issues
// {4x TDM load partner -> LDS, 1x 32KB TDM store [anchor|partner]x4 -> out},
// double-buffered and paced with s_wait_tensorcnt. Stores use TH=NT so the
// write-once output doesn't sweep L2.
// ---------------------------------------------------------------------------

namespace {
constexpr int kN = 1024;
constexpr int kC = 32;
constexpr int kK = 32;
constexpr int kF = 1024;
constexpr int kRowsPerAnchor = kK + kC - 1;        // 63
constexpr int kRowBytes      = kF * 4;             // 4096  (one source row)
constexpr int kOutRowBytes   = 2 * kF * 4;         // 8192  (one output row)
constexpr int kRowsPerStore  = 4;                  // rows batched per store
constexpr int kSlotBytes     = kRowsPerStore * kOutRowBytes; // 32768
constexpr int kNumSlots      = 2;                  // double buffer
}

typedef __attribute__((ext_vector_type(4))) unsigned int u32x4;
typedef __attribute__((ext_vector_type(8))) int          i32x8;
typedef __attribute__((ext_vector_type(4))) int          i32x4;

__device__ __forceinline__ unsigned mix_hash(unsigned a, unsigned b) {
  unsigned x = a * 0x9E3779B9u + (b ^ 0x7F4A7C15u) * 0x85EBCA6Bu;
  x ^= x >> 16; x *= 0x85EBCA6Bu;
  x ^= x >> 13; x *= 0xC2B2AE35u;
  x ^= x >> 16;
  return x;
}

#if defined(__gfx1250__)
// Build a 1-D tile descriptor (D# groups 0 and 1) per CDNA5 ISA.
__device__ __forceinline__ void tdm_make_desc(u32x4 &g0, i32x8 &g1,
                                              unsigned lds_addr,
                                              unsigned long long gaddr,
                                              unsigned nelem /* f32 elements */) {
  g0.x = 1u;                                    // count=1 (valid user descriptor)
  g0.y = lds_addr;                              // LDS byte address
  g0.z = (unsigned)(gaddr & 0xFFFFFFFFull);     // global_addr[31:0]
  g0.w = (unsigned)((gaddr >> 32) & 0x01FFFFFFull) | (2u << 30); // addr[56:32] | type=2

  g1.s0 = (int)(2u << 16);                      // data_size=2 (4B), mask=0, no opts
  g1.s1 = (int)((nelem & 0xFFFFu) << 16);       // tensor_dim0[15:0] (barrier addr = 0)
  g1.s2 = (int)((nelem >> 16) & 0xFFFFu);       // tensor_dim0[31:16]; tensor_dim1=0
  g1.s3 = (int)((nelem & 0xFFFFu) << 16);       // tile_dim0 = nelem (1-D tile)
  g1.s4 = 0;                                    // tile_dim1 = tile_dim2 = 0 (unused)
  g1.s5 = (int)nelem;                           // tensor_dim0_stride[31:0]
  g1.s6 = 0;                                    // stride hi / dim1_stride lo
  g1.s7 = 0;                                    // dim1_stride hi
}
#endif

template <int CPOL>
__device__ __forceinline__ void tdm_load_to_lds(unsigned lds_addr,
                                                const void *gsrc,
                                                unsigned nelem) {
#if defined(__gfx1250__)
  u32x4 g0; i32x8 g1;
  tdm_make_desc(g0, g1, lds_addr, (unsigned long long)(uintptr_t)gsrc, nelem);
  i32x4 z4 = {};
#if defined(__clang_major__) && (__clang_major__ >= 23)
  i32x8 z8 = {};
  __builtin_amdgcn_tensor_load_to_lds(g0, g1, z4, z4, z8, CPOL);
#else
  __builtin_amdgcn_tensor_load_to_lds(g0, g1, z4, z4, CPOL);
#endif
#endif
}

template <int CPOL>
__device__ __forceinline__ void tdm_store_from_lds(unsigned lds_addr,
                                                   void *gdst,
                                                   unsigned nelem) {
#if defined(__gfx1250__)
  u32x4 g0; i32x8 g1;
  tdm_make_desc(g0, g1, lds_addr, (unsigned long long)(uintptr_t)gdst, nelem);
  i32x4 z4 = {};
#if defined(__clang_major__) && (__clang_major__ >= 23)
  i32x8 z8 = {};
  __builtin_amdgcn_tensor_store_from_lds(g0, g1, z4, z4, z8, CPOL);
#else
  __builtin_amdgcn_tensor_store_from_lds(g0, g1, z4, z4, CPOL);
#endif
#endif
}

template <short N>
__device__ __forceinline__ void wait_tensorcnt() {
#if defined(__gfx1250__)
  __builtin_amdgcn_s_wait_tensorcnt(N);
#endif
}

__global__ __launch_bounds__(32)
void ExpandedSiameseConcat_tdm(const float* __restrict__ curr,
                               const int*   __restrict__ targets,
                               float*       __restrict__ out) {
  // Two 32KB slots, each holding 4 output rows: [anchor|partner] x 4.
  __shared__ __align__(128) unsigned char lds_buf[kNumSlots * kSlotBytes];

  const int i   = blockIdx.x;           // anchor row 0..1023
  const int lane = threadIdx.x;
  const int cls = targets[i];           // == i / kK for this dataset

  float* out_targets  = out;                                   // 64512 labels
  float* out_expanded = out + (size_t)kN * kRowsPerAnchor;     // 64512 x 2048

  // Labels: 1.0 for the K positive slots, 0.0 for the C-1 negative slots.
  for (int j = lane; j < kRowsPerAnchor; j += 32) {
    out_targets[(size_t)i * kRowsPerAnchor + j] = (j < kK) ? 1.0f : 0.0f;
  }

  // Flat LDS address: aperture lives in bits [63:32], offset is bits [31:0].
  const unsigned lds0 = (unsigned)(uintptr_t)(void*)&lds_buf[0];
  const float*  anchor = curr + (size_t)i * kF;

  // Stage the anchor row into the first half of every row position of both
  // slots (done once; reused by all 63 output rows of this anchor).
  for (int s = 0; s < kNumSlots; ++s)
    for (int r = 0; r < kRowsPerStore; ++r)
      tdm_load_to_lds<0>(lds0 + (unsigned)(s * kSlotBytes + r * kOutRowBytes),
                         anchor, kF);

  for (int jb = 0; jb < kRowsPerAnchor; jb += kRowsPerStore) {
    const int nrows = (kRowsPerAnchor - jb) < kRowsPerStore
                          ? (kRowsPerAnchor - jb) : kRowsPerStore;
    const unsigned slot = (unsigned)((jb / kRowsPerStore) & (kNumSlots - 1))
                          * (unsigned)kSlotBytes;

    // Keep the previous iteration's {4 loads, 1 store} in flight, but
    // guarantee the store that last read this slot (2 iterations back) is
    // done before we overwrite it.
    wait_tensorcnt<5>();

    for (int r = 0; r < nrows; ++r) {
      const int j = jb + r;
      int src;
      if (j < kK) {
        src = cls * kK + j;                            // positive: class member j
      } else {
        const int jn = j - kK;
        const int c  = jn + (jn >= cls ? 1 : 0);       // skip own class
        const int m  = 1 + (int)(mix_hash((unsigned)i, (unsigned)jn)
                                 % (unsigned)(kK - 1)); // member in [1, K)
        src = c * kK + m;                              // sampled negative
      }
      // Partner row -> second half of row position r in this slot.
      tdm_load_to_lds<0>(lds0 + slot + (unsigned)(r * kOutRowBytes + kRowBytes),
                         curr + (size_t)src * kF, kF);
    }

    // One batched store: nrows contiguous output rows (up to 32KB), TH=NT
    // (write-once stream; don't sweep L2).
    tdm_store_from_lds<1>(lds0 + slot,
                          out_expanded
                            + ((size_t)i * kRowsPerAnchor + jb) * (size_t)(2 * kF),
                          (unsigned)(nrows * 2 * kF));
  }

  wait_tensorcnt<0>();
}

extern "C" void kernel_launch(void* const* d_in, const int* in_sizes, int n_in,
                              void* d_out, int out_size, void* d_ws, size_t ws_size,
                              hipStream_t stream) {
  (void)in_sizes; (void)n_in; (void)out_size; (void)d_ws; (void)ws_size;
  const float* curr    = (const float*)d_in[0];
  const int*   targets = (const int*)d_in[1];
  float*       out     = (float*)d_out;
  hipLaunchKernelGGL(ExpandedSiameseConcat_tdm,
                     dim3(kN), dim3(32), 0, stream,
                     curr, targets, out);
}